// SparseConv2D_23313082482822
// MI455X (gfx1250) — compile-verified
//
#include <hip/hip_runtime.h>

// Sparse 3x3 conv (COO weights, 0.35% dense) for MI455X / gfx1250.
// - V_WMMA_F32_16X16X4_F32 as a gather->channel-routing FMA engine
//   (B is a one-hot-scaled 4x16 selector; C = 16 pixels x 16 out-channels).
// - Tensor Data Mover stages the input tile into LDS, using TDM pad_interval/
//   pad_amount to produce the bank-conflict-free 260-dword pixel stride and
//   tensor_dim-based OOB zero-fill for edge tiles.

#define C_CH      256
#define O_CH      256
#define H_IN      128
#define W_IN      128
#define H_OUT     126
#define W_OUT     126
#define BATCH     4

#define TILE_W    8
#define TILE_H    4
#define LDSW      (TILE_W + 2)        // 10 input cols
#define LDSH      (TILE_H + 2)        // 6 input rows
#define PIXSTRIDE 260                 // 256 ch + 4 pad dwords (TDM pad reproduces this)
#define NSLOTS    (LDSW * LDSH)       // 60 input pixels staged per block

#define NBUCKET   16                  // output channels grouped 16-wide
#define MAXE      512                 // cap per bucket (E[n]=128, >20 sigma headroom)

typedef __attribute__((ext_vector_type(2))) float        v2f;
typedef __attribute__((ext_vector_type(8))) float        v8f;
typedef __attribute__((ext_vector_type(4))) unsigned int v4u;
typedef __attribute__((ext_vector_type(4))) int          v4i;
typedef __attribute__((ext_vector_type(8))) int          v8i;

#if defined(__has_builtin)
#  if __has_builtin(__builtin_amdgcn_tensor_load_to_lds)
#    define USE_TDM 1
#  endif
#endif

// ---------------------------------------------------------------------------
// Phase 1: deterministic COO -> per-ocblock buckets. Single wave; lane==bucket
// appends in nnz order (no atomics => bit-identical across graph replays).
// Applies the reference's (kh,kw,C) vs (C,kh,kw) flattening remap:
//   k = i*768 + j*256 + c ; c_u = k/9 ; i_u = (k%9)/3 ; j_u = k%3
// Packs entry = { ldsGatherOffset | (col<<16), f32 value }.
// ---------------------------------------------------------------------------
__global__ void sparse_bucket_build(const long long* __restrict__ wi,
                                    const float* __restrict__ wv,
                                    int nnz,
                                    int* __restrict__ cnts,
                                    int2* __restrict__ ent) {
    int lane = threadIdx.x;
    int myCnt = 0;
    for (int n = 0; n < nnz; ++n) {
        int o = (int)wi[4 * n + 0];
        int i = (int)wi[4 * n + 1];
        int j = (int)wi[4 * n + 2];
        int c = (int)wi[4 * n + 3];
        int k  = i * 768 + j * 256 + c;
        int cu = k / 9;
        int tt = k - cu * 9;
        int iu = tt / 3;
        int ju = tt - iu * 3;
        int goff = (iu * LDSW + ju) * PIXSTRIDE + cu;   // < 2^14, fits 16 bits
        int ocb = o >> 4;
        int col = o & 15;
        float v = wv[n];
        if (lane == ocb && myCnt < MAXE) {
            ent[ocb * MAXE + myCnt] = make_int2(goff | (col << 16), __float_as_int(v));
            ++myCnt;
        }
    }
    if (lane < NBUCKET) {
        while ((myCnt & 3) && myCnt < MAXE) {           // pad to K-group of 4
            ent[lane * MAXE + myCnt] = make_int2(0, 0); // value 0.0 => no-op MAC
            ++myCnt;
        }
        cnts[lane] = myCnt;
    }
}

// ---------------------------------------------------------------------------
// Phase 2: main kernel. 64 threads = 2 waves per block; each wave owns an
// 8x2 pixel patch (M=16) and sweeps 16 output-channel blocks (N=16),
// accumulating with v_wmma_f32_16x16x4_f32 over 4-nnz K-groups.
// ---------------------------------------------------------------------------
__launch_bounds__(64)
__global__ void sparse_conv_wmma(const float* __restrict__ in,
                                 const int*  __restrict__ cnts,
                                 const int2* __restrict__ ent,
                                 float* __restrict__ out) {
    __shared__ float tile[NSLOTS * PIXSTRIDE];          // 62,400 B

    const int t  = threadIdx.x;
    const int x0 = blockIdx.x * TILE_W;
    const int y0 = blockIdx.y * TILE_H;
    const int b  = blockIdx.z;
    const int wave = t >> 5;

#ifdef USE_TDM
    // ---- Tensor Data Mover fill: one DMA descriptor per block -------------
    // Tile = 2D slab: dim0 = LDSW*256 floats (pixels*channels within a row),
    // dim1 = LDSH rows, row stride = W_IN*256 floats. pad_interval=256 dwords
    // + pad_amount=4 dwords => LDS pixel stride 260. tensor_dim = remaining
    // extent from tile start => TDM zero-fills out-of-image reads.
    if (wave == 0) {
        unsigned long long gaddr =
            (unsigned long long)(const void*)
            (in + ((((long long)b * H_IN + y0) * W_IN + x0) * C_CH));
        unsigned int lds_base = (unsigned int)(unsigned long long)&tile[0];

        unsigned int td0 = (unsigned int)((W_IN - x0) * C_CH);  // floats valid in dim0
        unsigned int td1 = (unsigned int)(H_IN - y0);           // rows valid in dim1
        const unsigned int tile_d0 = LDSW * C_CH;               // 2560 (fits 16b)
        const unsigned int tile_d1 = LDSH;                      // 6
        const unsigned long long s0 = (unsigned long long)W_IN * C_CH; // 32768

        v4u g0 = { 1u,                                          // count=1, user mode
                   lds_base,                                    // lds_addr (bytes)
                   (unsigned int)(gaddr & 0xFFFFFFFFull),       // global_addr lo
                   (unsigned int)((gaddr >> 32) & 0x1FFFFFFull) // global_addr hi (57b)
                       | (2u << 30) };                          // type=2 ("image")

        v8i g1 = { (int)((2u << 16)            // data_size = 4B
                       | (1u << 20)            // pad_enable
                       | (7u << 22)            // pad_interval: 256 dwords
                       | (3u << 25)),          // pad_amount:   4 dwords
                   (int)((td0 & 0xFFFFu) << 16),                         // dim0[15:0]
                   (int)(((td0 >> 16) & 0xFFFFu) | ((td1 & 0xFFFFu) << 16)),
                   (int)(((td1 >> 16) & 0xFFFFu) | (tile_d0 << 16)),     // tile_dim0
                   (int)tile_d1,                                          // tile_dim1
                   (int)(s0 & 0xFFFFFFFFull),                             // stride0 lo
                   (int)((s0 >> 32) & 0xFFFFull),                         // stride0 hi
                   0 };                                                   // stride1 = 0

        v4i gz = { 0, 0, 0, 0 };
#if __clang_major__ >= 23
        v8i gz8 = { 0, 0, 0, 0, 0, 0, 0, 0 };
        __builtin_amdgcn_tensor_load_to_lds(g0, g1, gz, gz, gz8, 0);
#else
        __builtin_amdgcn_tensor_load_to_lds(g0, g1, gz, gz, 0);
#endif
        __builtin_amdgcn_s_wait_tensorcnt(0);
    }
    __syncthreads();
#else
    // ---- Fallback cooperative fill (plain b128 loads + LDS stores) --------
    for (int idx = t; idx < NSLOTS * 64; idx += 64) {
        int slot = idx >> 6;
        int ch4  = (idx & 63) << 2;
        int r    = slot / LDSW;
        int cc   = slot - r * LDSW;
        int py   = y0 + r;
        int px   = x0 + cc;
        float4 v = make_float4(0.f, 0.f, 0.f, 0.f);
        if (py < H_IN && px < W_IN) {
            v = *(const float4*)(in + ((((long long)b * H_IN + py) * W_IN + px) * C_CH + ch4));
        }
        *(float4*)(&tile[slot * PIXSTRIDE + ch4]) = v;
    }
    __syncthreads();
#endif

    const int  lane = t & 31;
    const int  ln   = lane & 15;        // pixel index M (A) / column N (B,C)
    const bool hi   = lane >= 16;       // high half handles K=2,3
    const int  rowl = 2 * wave + (ln >> 3);
    const int  coll = ln & 7;
    const int  pixBase = (rowl * LDSW + coll) * PIXSTRIDE;

    for (int ocb = 0; ocb < NBUCKET; ++ocb) {
        const int  cnt = cnts[ocb];
        const int2* e  = ent + ocb * MAXE;
        v8f acc = {};
        for (int g = 0; g < cnt; g += 4) {
            __builtin_prefetch((const void*)(e + g + 8), 0, 1);
            int4 p0 = *(const int4*)(e + g);        // entries g, g+1
            int4 p1 = *(const int4*)(e + g + 2);    // entries g+2, g+3
            int g0 = p0.x & 0xFFFF, c0 = (p0.x >> 16) & 15;
            int g1 = p0.z & 0xFFFF, c1 = (p0.z >> 16) & 15;
            int g2 = p1.x & 0xFFFF, c2 = (p1.x >> 16) & 15;
            int g3 = p1.z & 0xFFFF, c3 = (p1.z >> 16) & 15;
            float w0 = __int_as_float(p0.y);
            float w1 = __int_as_float(p0.w);
            float w2 = __int_as_float(p1.y);
            float w3 = __int_as_float(p1.w);

            // A: 16x4 f32 — lane holds its pixel; VGPR0=K0|K2, VGPR1=K1|K3.
            v2f A;
            A.x = tile[pixBase + (hi ? g2 : g0)];   // per-lane LDS gather
            A.y = tile[pixBase + (hi ? g3 : g1)];

            // B: 4x16 one-hot-scaled selector; VGPR0=K0|K2 rows, VGPR1=K1|K3.
            v2f B;
            B.x = hi ? (ln == c2 ? w2 : 0.f) : (ln == c0 ? w0 : 0.f);
            B.y = hi ? (ln == c3 ? w3 : 0.f) : (ln == c1 ? w1 : 0.f);

            acc = __builtin_amdgcn_wmma_f32_16x16x4_f32(
                false, A, false, B, (short)0, acc, false, false);
        }

        // C layout: VGPR r -> M=r (lanes 0-15) / M=8+r (lanes 16-31), N=ln.
        #pragma unroll
        for (int r = 0; r < 8; ++r) {
            int m  = hi ? (8 + r) : r;
            int oy = y0 + 2 * wave + (m >> 3);
            int ox = x0 + (m & 7);
            if (oy < H_OUT && ox < W_OUT) {
                out[(((long long)b * H_OUT + oy) * W_OUT + ox) * O_CH + (ocb << 4) + ln]
                    = acc[r];
            }
        }
    }
}

// ---------------------------------------------------------------------------
extern "C" void kernel_launch(void* const* d_in, const int* in_sizes, int n_in,
                              void* d_out, int out_size, void* d_ws, size_t ws_size,
                              hipStream_t stream) {
    const float*     in = (const float*)d_in[0];
    const long long* wi = (const long long*)d_in[1];   // int64 COO indices [nnz,4]
    const float*     wv = (const float*)d_in[2];
    const int nnz = in_sizes[2];

    int*  cnts = (int*)d_ws;
    int2* ent  = (int2*)((char*)d_ws + 64);            // 16*512*8B = 64KB buckets

    sparse_bucket_build<<<1, 32, 0, stream>>>(wi, wv, nnz, cnts, ent);

    dim3 grid((W_OUT + TILE_W - 1) / TILE_W,           // 16
              (H_OUT + TILE_H - 1) / TILE_H,           // 32
              BATCH);                                  // 4
    sparse_conv_wmma<<<grid, 64, 0, stream>>>(in, cnts, ent, (float*)d_out);
}